// GNN_23656679866485
// MI455X (gfx1250) — compile-verified
//
#include <hip/hip_runtime.h>
#include <hip/hip_bf16.h>

#define N_NODES 50000
#define N_EDGES 800000
#define D_FEAT  128
#define HIDDEN  128
#define OUT_C   64

typedef __attribute__((ext_vector_type(2))) float v2f;
typedef __attribute__((ext_vector_type(8))) float v8f;

// ---------------------------------------------------------------------------
// Scatter-mean accumulation: one wave (32 lanes) per edge, 4 floats per lane.
// agg[dst] += x[src]; deg[dst] += 1 (first pass only).
// ---------------------------------------------------------------------------
__global__ void GNN_scatter_kernel(const float* __restrict__ x,
                                   const long long* __restrict__ ei,
                                   float* __restrict__ agg,
                                   float* __restrict__ deg,
                                   int E, int add_deg) {
    int idx  = blockIdx.x * blockDim.x + threadIdx.x;
    int e    = idx >> 5;
    int lane = idx & 31;
    if (e >= E) return;
    int s = (int)ei[e];
    int d = (int)ei[E + e];
    const float4* xs = (const float4*)(x + (size_t)s * D_FEAT);
    float4 v = xs[lane];
    float* ap = agg + (size_t)d * D_FEAT + lane * 4;
    atomicAdd(ap + 0, v.x);
    atomicAdd(ap + 1, v.y);
    atomicAdd(ap + 2, v.z);
    atomicAdd(ap + 3, v.w);
    if (add_deg && lane == 0) atomicAdd(deg + d, 1.0f);
}

// ---------------------------------------------------------------------------
// SAGEConv via f32 WMMA:  out = act((agg/deg) @ Wl + b + xin @ Wr) [* mask]
// One 16-row node tile per block; wave w handles output columns [16w,16w+16).
// A fragment (16x4 f32): lane 16h+m holds A[m][k0+2h], A[m][k0+2h+1].
// B fragment (4x16 f32): lane 16h+n holds B[k0+2h][n], B[k0+2h+1][n].
// C/D (16x16 f32): acc[j] on lane 16h+n = C[j+8h][n].
// ---------------------------------------------------------------------------
__global__ void GNN_sage_wmma_kernel(const float* __restrict__ agg,
                                     const float* __restrict__ deg,
                                     const float* __restrict__ xin,
                                     const float* __restrict__ Wl,
                                     const float* __restrict__ Wr,
                                     const float* __restrict__ bias,
                                     const float* __restrict__ mask,
                                     float* __restrict__ out,
                                     int K, int Ncols, int act) {
    const int wave = threadIdx.x >> 5;
    const int lane = threadIdx.x & 31;
    const int n    = lane & 15;       // col within tile / row within A tile
    const int hh   = lane >> 4;       // half selector
    const int row0 = blockIdx.x * 16;
    const int col0 = wave * 16;
    const int kofs = 2 * hh;

    const float* arow = agg + (size_t)(row0 + n) * K;
    const float* xrow = xin + (size_t)(row0 + n) * K;
    const float  rdeg = 1.0f / fmaxf(deg[row0 + n], 1.0f);

    v8f acc = {};

    // pass 1: mean aggregation @ Wl
    for (int kk = 0; kk < K; kk += 4) {
        v2f a, b;
        a[0] = arow[kk + kofs]     * rdeg;
        a[1] = arow[kk + kofs + 1] * rdeg;
        b[0] = Wl[(size_t)(kk + kofs)     * Ncols + col0 + n];
        b[1] = Wl[(size_t)(kk + kofs + 1) * Ncols + col0 + n];
        acc = __builtin_amdgcn_wmma_f32_16x16x4_f32(false, a, false, b,
                                                    (short)0, acc, false, false);
    }
    // pass 2: self features @ Wr
    for (int kk = 0; kk < K; kk += 4) {
        v2f a, b;
        a[0] = xrow[kk + kofs];
        a[1] = xrow[kk + kofs + 1];
        b[0] = Wr[(size_t)(kk + kofs)     * Ncols + col0 + n];
        b[1] = Wr[(size_t)(kk + kofs + 1) * Ncols + col0 + n];
        acc = __builtin_amdgcn_wmma_f32_16x16x4_f32(false, a, false, b,
                                                    (short)0, acc, false, false);
    }

    const float bc = bias[col0 + n];
#pragma unroll
    for (int j = 0; j < 8; ++j) {
        int row = row0 + j + 8 * hh;
        float v = acc[j] + bc;
        if (act) v = fmaxf(v, 0.0f) * mask[(size_t)row * Ncols + col0 + n];
        out[(size_t)row * Ncols + col0 + n] = v;
    }
}

// ---------------------------------------------------------------------------
// Row-wise log_softmax over 64 columns, in place. One wave per row.
// ---------------------------------------------------------------------------
__global__ void GNN_logsoftmax_kernel(float* __restrict__ out, int N) {
    int wave = threadIdx.x >> 5;
    int lane = threadIdx.x & 31;
    int row  = blockIdx.x * (blockDim.x >> 5) + wave;
    if (row >= N) return;
    float* p = out + (size_t)row * OUT_C;
    float x0 = p[lane];
    float x1 = p[lane + 32];
    float mx = fmaxf(x0, x1);
    for (int o = 16; o > 0; o >>= 1) mx = fmaxf(mx, __shfl_xor(mx, o, 32));
    float s = expf(x0 - mx) + expf(x1 - mx);
    for (int o = 16; o > 0; o >>= 1) s += __shfl_xor(s, o, 32);
    float l = mx + logf(s);
    p[lane]      = x0 - l;
    p[lane + 32] = x1 - l;
}

extern "C" void kernel_launch(void* const* d_in, const int* in_sizes, int n_in,
                              void* d_out, int out_size, void* d_ws, size_t ws_size,
                              hipStream_t stream) {
    const float*     x    = (const float*)d_in[0];
    const long long* ei   = (const long long*)d_in[1];   // int64 [2, E]
    const float*     mask = (const float*)d_in[2];
    const float*     W1l  = (const float*)d_in[3];
    const float*     b1   = (const float*)d_in[4];
    const float*     W1r  = (const float*)d_in[5];
    const float*     W2l  = (const float*)d_in[6];
    const float*     b2   = (const float*)d_in[7];
    const float*     W2r  = (const float*)d_in[8];
    float* out = (float*)d_out;

    float* agg = (float*)d_ws;                               // [N, 128]
    float* deg = agg + (size_t)N_NODES * HIDDEN;             // [N]
    float* h   = deg + N_NODES;                              // [N, 128]

    const int scat_blocks = (N_EDGES * 32) / 256;            // 100000
    const int tile_blocks = N_NODES / 16;                    // 3125

    // ---- layer 1 ----
    hipMemsetAsync(agg, 0, ((size_t)N_NODES * HIDDEN + N_NODES) * sizeof(float), stream);
    GNN_scatter_kernel<<<scat_blocks, 256, 0, stream>>>(x, ei, agg, deg, N_EDGES, 1);
    GNN_sage_wmma_kernel<<<tile_blocks, 8 * 32, 0, stream>>>(
        agg, deg, x, W1l, W1r, b1, mask, h, D_FEAT, HIDDEN, 1);

    // ---- layer 2 ----
    hipMemsetAsync(agg, 0, (size_t)N_NODES * HIDDEN * sizeof(float), stream);
    GNN_scatter_kernel<<<scat_blocks, 256, 0, stream>>>(h, ei, agg, deg, N_EDGES, 0);
    GNN_sage_wmma_kernel<<<tile_blocks, 4 * 32, 0, stream>>>(
        agg, deg, h, W2l, W2r, b2, nullptr, out, HIDDEN, OUT_C, 0);

    // ---- log_softmax ----
    GNN_logsoftmax_kernel<<<(N_NODES + 7) / 8, 256, 0, stream>>>(out, N_NODES);
}